// GCN_ancestor_38981123179103
// MI455X (gfx1250) — compile-verified
//
#include <hip/hip_runtime.h>
#include <hip/hip_bf16.h>
#include <math.h>

typedef __attribute__((ext_vector_type(2))) float v2f;
typedef __attribute__((ext_vector_type(8))) float v8f;

#define TPB 256

// ---------------- degree / normalization ----------------
__global__ void k_init_deg(float* __restrict__ deg, int N) {
    int i = blockIdx.x * blockDim.x + threadIdx.x;
    if (i < N) deg[i] = 1.0f;                       // self-loop contribution
}

__global__ void k_accum_deg(const int* __restrict__ dst, float* __restrict__ deg, int E) {
    int e = blockIdx.x * blockDim.x + threadIdx.x;
    if (e < E) atomicAdd(&deg[dst[e]], 1.0f);
}

__global__ void k_deg_to_dinv(float* __restrict__ deg, int N) {
    int i = blockIdx.x * blockDim.x + threadIdx.x;
    if (i < N) deg[i] = rsqrtf(deg[i]);             // deg >= 1 always (self loop)
}

// ---------------- GEMM1: h1[N,16] = x2[N,128] @ W1[128,16] via WMMA f32 16x16x4 ----------------
__global__ void k_gemm1_wmma(const float* __restrict__ X, const float* __restrict__ W,
                             float* __restrict__ H, int N) {
    int wave = blockIdx.x * (blockDim.x >> 5) + (threadIdx.x >> 5);
    int lane = threadIdx.x & 31;
    int row0 = wave << 4;
    if (row0 >= N) return;                           // wave-uniform: EXEC stays all-ones
    int half = lane >> 4;                            // 0: K offsets {0,1}; 1: {2,3}
    int l16  = lane & 15;
    int arow = row0 + l16; if (arow >= N) arow = N - 1;
    const float* xr = X + (size_t)arow * 128;

    v8f c = {};
#pragma unroll
    for (int k0 = 0; k0 < 128; k0 += 4) {
        int ka = k0 + half * 2;
        v2f a; a.x = xr[ka];             a.y = xr[ka + 1];
        v2f b; b.x = W[ka * 16 + l16];   b.y = W[(ka + 1) * 16 + l16];
        c = __builtin_amdgcn_wmma_f32_16x16x4_f32(false, a, false, b,
                                                  (short)0, c, false, false);
    }
#pragma unroll
    for (int i = 0; i < 8; ++i) {
        int r = row0 + i + half * 8;
        if (r < N) H[(size_t)r * 16 + l16] = c[i];
    }
}

// ---------------- GEMM2: h3[N,10] = h2[N,16] @ Wend[16,10] via WMMA (B padded to 16 cols) ----------------
__global__ void k_gemm2_wmma(const float* __restrict__ X, const float* __restrict__ W,
                             float* __restrict__ H, int N) {
    int wave = blockIdx.x * (blockDim.x >> 5) + (threadIdx.x >> 5);
    int lane = threadIdx.x & 31;
    int row0 = wave << 4;
    if (row0 >= N) return;
    int half = lane >> 4;
    int l16  = lane & 15;
    int arow = row0 + l16; if (arow >= N) arow = N - 1;
    const float* xr = X + (size_t)arow * 16;

    v8f c = {};
#pragma unroll
    for (int k0 = 0; k0 < 16; k0 += 4) {
        int ka = k0 + half * 2;
        v2f a; a.x = xr[ka]; a.y = xr[ka + 1];
        v2f b;
        b.x = (l16 < 10) ? W[ka * 10 + l16]       : 0.0f;   // select, not branch: EXEC unchanged
        b.y = (l16 < 10) ? W[(ka + 1) * 10 + l16] : 0.0f;
        c = __builtin_amdgcn_wmma_f32_16x16x4_f32(false, a, false, b,
                                                  (short)0, c, false, false);
    }
    if (l16 < 10) {
#pragma unroll
        for (int i = 0; i < 8; ++i) {
            int r = row0 + i + half * 8;
            if (r < N) H[(size_t)r * 10 + l16] = c[i];
        }
    }
}

// ---------------- propagation: init with self-loop term, then edge scatter ----------------
__global__ void k_selfloop_init(const float* __restrict__ Hs, const float* __restrict__ dinv,
                                float* __restrict__ P, long long total, int Fdim) {
    long long t = (long long)blockIdx.x * blockDim.x + threadIdx.x;
    if (t >= total) return;
    int i = (int)(t / Fdim);
    float di = dinv[i];
    P[t] = Hs[t] * di * di;
}

__global__ void k_edge_scatter(const int* __restrict__ src, const int* __restrict__ dst,
                               const float* __restrict__ Hs, const float* __restrict__ dinv,
                               float* __restrict__ P, int E, int Fdim) {
    int t = blockIdx.x * blockDim.x + threadIdx.x;
    int e = t >> 4;           // 16 lanes per edge
    int f = t & 15;
    if (e >= E || f >= Fdim) return;
    int s = src[e], d = dst[e];
    float nrm = dinv[s] * dinv[d];
    atomicAdd(&P[(size_t)d * Fdim + f], Hs[(size_t)s * Fdim + f] * nrm);
}

__global__ void k_bias_relu(float* __restrict__ P, const float* __restrict__ b,
                            long long total, int Fdim) {
    long long t = (long long)blockIdx.x * blockDim.x + threadIdx.x;
    if (t >= total) return;
    int f = (int)(t % Fdim);
    float v = P[t] + b[f];
    P[t] = v > 0.0f ? v : 0.0f;
}

// ---------------- final: +b_end, log_softmax over C=10 ----------------
__global__ void k_logsoftmax(const float* __restrict__ P, const float* __restrict__ b,
                             float* __restrict__ out, int N) {
    int i = blockIdx.x * blockDim.x + threadIdx.x;
    if (i >= N) return;
    float v[10];
    float m = -INFINITY;
#pragma unroll
    for (int f = 0; f < 10; ++f) {
        v[f] = P[(size_t)i * 10 + f] + b[f];
        m = fmaxf(m, v[f]);
    }
    float s = 0.0f;
#pragma unroll
    for (int f = 0; f < 10; ++f) s += expf(v[f] - m);
    float lse = m + logf(s);
#pragma unroll
    for (int f = 0; f < 10; ++f) out[(size_t)i * 10 + f] = v[f] - lse;
}

// ---------------- launcher ----------------
extern "C" void kernel_launch(void* const* d_in, const int* in_sizes, int n_in,
                              void* d_out, int out_size, void* d_ws, size_t ws_size,
                              hipStream_t stream) {
    // Inputs: 0:x1 1:x2 2:W1 3:b1 4:W_end 5:b_end 6:edge_index1 7:edge_index2 8:skip
    // The x1/edge_index1 branch never reaches the returned value -> skipped entirely.
    const float* x2   = (const float*)d_in[1];
    const float* W1   = (const float*)d_in[2];
    const float* b1   = (const float*)d_in[3];
    const float* Wend = (const float*)d_in[4];
    const float* bend = (const float*)d_in[5];
    const int*   ei2  = (const int*)  d_in[7];

    const int N = in_sizes[1] / 128;
    const int E = in_sizes[7] / 2;
    const int* src = ei2;        // edge_index[0]
    const int* dst = ei2 + E;    // edge_index[1]

    char*  ws  = (char*)d_ws;
    size_t off = 0;
    auto take = [&](size_t elems) -> float* {
        float* p = (float*)(ws + off);
        off += (elems * sizeof(float) + 255) & ~(size_t)255;
        return p;
    };
    float* dinv = take((size_t)N);        // deg -> dinv in place
    float* h1   = take((size_t)N * 16);   // x2 @ W1
    float* p1   = take((size_t)N * 16);   // propagated, then relu'd h2 (in place)
    float* h3   = take((size_t)N * 10);   // h2 @ W_end
    float* p2   = take((size_t)N * 10);   // propagated logits (pre-bias)
    (void)ws_size; (void)n_in; (void)out_size;

    const int B = TPB;
    const int wpb = B / 32;
    const int tiles = (N + 15) / 16;
    const long long t16 = (long long)N * 16;
    const long long t10 = (long long)N * 10;
    const long long eT  = (long long)E * 16;

    // degrees (shared by both convs on edge_index2)
    k_init_deg   <<<(N + B - 1) / B, B, 0, stream>>>(dinv, N);
    k_accum_deg  <<<(E + B - 1) / B, B, 0, stream>>>(dst, dinv, E);
    k_deg_to_dinv<<<(N + B - 1) / B, B, 0, stream>>>(dinv, N);

    // conv1 on x2
    k_gemm1_wmma   <<<(tiles + wpb - 1) / wpb, B, 0, stream>>>(x2, W1, h1, N);
    k_selfloop_init<<<(int)((t16 + B - 1) / B), B, 0, stream>>>(h1, dinv, p1, t16, 16);
    k_edge_scatter <<<(int)((eT + B - 1) / B), B, 0, stream>>>(src, dst, h1, dinv, p1, E, 16);
    k_bias_relu    <<<(int)((t16 + B - 1) / B), B, 0, stream>>>(p1, b1, t16, 16);

    // conv2 on h2 (= p1)
    k_gemm2_wmma   <<<(tiles + wpb - 1) / wpb, B, 0, stream>>>(p1, Wend, h3, N);
    k_selfloop_init<<<(int)((t10 + B - 1) / B), B, 0, stream>>>(h3, dinv, p2, t10, 10);
    k_edge_scatter <<<(int)((eT + B - 1) / B), B, 0, stream>>>(src, dst, h3, dinv, p2, E, 10);

    // bias + log_softmax
    k_logsoftmax   <<<(N + B - 1) / B, B, 0, stream>>>(p2, bend, (float*)d_out, N);
}